// TverskyLoss_63299228009020
// MI455X (gfx1250) — compile-verified
//
#include <hip/hip_runtime.h>
#include <hip/hip_bf16.h>

// Tversky loss, MI455X (gfx1250).
// Memory-bound: 144 MB read once -> ~6.2 us floor at 23.3 TB/s.
// Strategy: streaming b128 loads, per-lane fp32 accumulators (24 per thread),
// exact-fp32 cross-lane reduction on the matrix pipe via V_WMMA_F32_16X16X4_F32
// (B = ones => row sums), deterministic block partials + tiny final kernel.

typedef __attribute__((ext_vector_type(2))) float v2f;
typedef __attribute__((ext_vector_type(8))) float v8f;

#define TVERSKY_EPS 1e-7f

constexpr int kC       = 8;
constexpr int kHW      = 512 * 512;
constexpr int kB       = 16;
constexpr int kBlocksX = 64;
constexpr int kTPB     = 256;

// Exact fp32 sum of x over all 32 lanes, broadcast to every lane.
// Stage 1: D1[m,n] = x[m] + x[m+16]            (A = (x,0), B = ones, C = 0)
// Stage 2: chain 4 WMMAs with A = (d1[2j],d1[2j+1]) accumulating through C:
//          every element of the result = sum over all 16 row-sums = full sum.
__device__ __forceinline__ float wave_sum_wmma(float x) {
  v2f a;    a.x = x;     a.y = 0.0f;
  v2f ones; ones.x = 1.0f; ones.y = 1.0f;
  v8f z = {};
  v8f d1 = __builtin_amdgcn_wmma_f32_16x16x4_f32(false, a, false, ones, (short)0, z, false, false);
  v8f acc = {};
  v2f p;
  p.x = d1[0]; p.y = d1[1];
  acc = __builtin_amdgcn_wmma_f32_16x16x4_f32(false, p, false, ones, (short)0, acc, false, false);
  p.x = d1[2]; p.y = d1[3];
  acc = __builtin_amdgcn_wmma_f32_16x16x4_f32(false, p, false, ones, (short)0, acc, false, false);
  p.x = d1[4]; p.y = d1[5];
  acc = __builtin_amdgcn_wmma_f32_16x16x4_f32(false, p, false, ones, (short)0, acc, false, false);
  p.x = d1[6]; p.y = d1[7];
  acc = __builtin_amdgcn_wmma_f32_16x16x4_f32(false, p, false, ones, (short)0, acc, false, false);
  return acc[0];
}

__device__ __forceinline__ float get_f4(const float4& v, int j) {
  return j == 0 ? v.x : (j == 1 ? v.y : (j == 2 ? v.z : v.w));
}
__device__ __forceinline__ int get_i4(const int4& v, int j) {
  return j == 0 ? v.x : (j == 1 ? v.y : (j == 2 ? v.z : v.w));
}

// Per-block partial stats: ws[(b*kBlocksX + bx)*24 + k],
// k = 0..7: sum softmax, 8..15: intersection, 16..23: target count.
__global__ void __launch_bounds__(kTPB)
tversky_main(const float* __restrict__ logits, const int* __restrict__ target,
             float* __restrict__ ws) {
  const int b   = blockIdx.y;
  const int tid = threadIdx.x;
  const float* base  = logits + (size_t)b * kC * kHW;
  const int*   tbase = target + (size_t)b * kHW;

  float sumP[kC], inter[kC], cnt[kC];
#pragma unroll
  for (int c = 0; c < kC; ++c) { sumP[c] = 0.0f; inter[c] = 0.0f; cnt[c] = 0.0f; }

  const int stride = kBlocksX * kTPB * 4;  // pixels per grid-stride step
  for (int p = (blockIdx.x * kTPB + tid) * 4; p < kHW; p += stride) {
    float4 x[kC];
#pragma unroll
    for (int c = 0; c < kC; ++c)
      x[c] = *(const float4*)(base + (size_t)c * kHW + p);
    const int4 t4 = *(const int4*)(tbase + p);

#pragma unroll
    for (int j = 0; j < 4; ++j) {
      float xv[kC];
#pragma unroll
      for (int c = 0; c < kC; ++c) xv[c] = get_f4(x[c], j);
      float m = xv[0];
#pragma unroll
      for (int c = 1; c < kC; ++c) m = fmaxf(m, xv[c]);
      float e[kC];
      float s = 0.0f;
#pragma unroll
      for (int c = 0; c < kC; ++c) { e[c] = __expf(xv[c] - m); s += e[c]; }
      const float inv = __builtin_amdgcn_rcpf(s);
      const int   t   = get_i4(t4, j);
#pragma unroll
      for (int c = 0; c < kC; ++c) {
        const float pc = e[c] * inv;
        const bool  hit = (t == c);
        sumP[c]  += pc;
        inter[c] += hit ? pc : 0.0f;
        cnt[c]   += hit ? 1.0f : 0.0f;
      }
    }
  }

  // --- block reduction: WMMA wave sums -> LDS -> one store per stat ---
  float vals[24];
#pragma unroll
  for (int c = 0; c < kC; ++c) {
    vals[c]      = sumP[c];
    vals[8 + c]  = inter[c];
    vals[16 + c] = cnt[c];
  }
  float wsum[24];
#pragma unroll
  for (int k = 0; k < 24; ++k) wsum[k] = wave_sum_wmma(vals[k]);  // EXEC all-ones here

  __shared__ float red[kTPB / 32][24];
  const int wave = tid >> 5;
  const int lane = tid & 31;
  if (lane == 0) {
#pragma unroll
    for (int k = 0; k < 24; ++k) red[wave][k] = wsum[k];
  }
  __syncthreads();
  if (tid < 24) {
    float s = red[0][tid];
#pragma unroll
    for (int w = 1; w < kTPB / 32; ++w) s += red[w][tid];
    ws[((size_t)b * kBlocksX + blockIdx.x) * 24 + tid] = s;
  }
}

// 128 threads: thread i handles (b,c) = (i>>3, i&7); deterministic block-order sums,
// then WMMA wave reductions of the 128 per-class losses -> mean.
__global__ void __launch_bounds__(128)
tversky_final(const float* __restrict__ ws, float* __restrict__ out) {
  const int tid = threadIdx.x;          // 0..127, all active
  const int b = tid >> 3, c = tid & 7;
  const float* base = ws + (size_t)b * kBlocksX * 24;
  float S = 0.0f, I = 0.0f, T = 0.0f;
  for (int blk = 0; blk < kBlocksX; ++blk) {
    const float* p = base + blk * 24;
    S += p[c];
    I += p[8 + c];
    T += p[16 + c];
  }
  const float fp = S - I;
  const float fn = T - I;
  const float tv = (I + TVERSKY_EPS) / (I + 0.5f * fp + 0.5f * fn + TVERSKY_EPS);
  const float loss = 1.0f - tv;

  const float wsum = wave_sum_wmma(loss);   // per-wave totals (EXEC all-ones)
  __shared__ float part[4];
  if ((tid & 31) == 0) part[tid >> 5] = wsum;
  __syncthreads();
  if (tid == 0)
    out[0] = (part[0] + part[1] + part[2] + part[3]) * (1.0f / 128.0f);
}

extern "C" void kernel_launch(void* const* d_in, const int* in_sizes, int n_in,
                              void* d_out, int out_size, void* d_ws, size_t ws_size,
                              hipStream_t stream) {
  (void)in_sizes; (void)n_in; (void)out_size; (void)ws_size;
  const float* logits = (const float*)d_in[0];
  const int*   target = (const int*)d_in[1];
  float* ws  = (float*)d_ws;   // needs 16*64*24*4 = 96 KB; fully overwritten each call
  float* out = (float*)d_out;

  dim3 grid(kBlocksX, kB);
  tversky_main<<<grid, kTPB, 0, stream>>>(logits, target, ws);
  tversky_final<<<1, 128, 0, stream>>>(ws, out);
}